// SpatialGatherModule_31834297598793
// MI455X (gfx1250) — compile-verified
//
#include <hip/hip_runtime.h>

// ---------------------------------------------------------------------------
// SpatialGather: per-segment softmax (K=20 classes) + weighted aggregation of
// point features (C=96) -> context [B=8, K=20, C=96].
// Memory-bound (~82MB traffic => ~3.5us floor @ 23.3TB/s). Contraction done
// with V_WMMA_F32_16X16X4_F32 (fp32 A/B/C keeps reference precision).
// ---------------------------------------------------------------------------

typedef __attribute__((ext_vector_type(2))) float v2f;
typedef __attribute__((ext_vector_type(8))) float v8f;

#define KCLS   20     // classes
#define CFEAT  96     // feature channels
#define BMAX   8      // batch segments
#define SSTEP  64     // points per LDS superstep
#define WSTR   40     // LDS row stride (dwords): 2*40 mod 64 = 16 -> no bank conflicts
#define SCALE_F 1.0f

// Monotonic float<->uint mapping for order-invariant (deterministic) atomic max.
__device__ __forceinline__ unsigned enc_f32(float f) {
  unsigned u = __float_as_uint(f);
  return (u & 0x80000000u) ? ~u : (u | 0x80000000u);
}
__device__ __forceinline__ float dec_f32(unsigned u) {
  unsigned b = (u >> 31) ? (u ^ 0x80000000u) : ~u;
  return __uint_as_float(b);
}
__device__ __forceinline__ bool finitef_bits(float x) {
  return (__float_as_uint(x) & 0x7F800000u) != 0x7F800000u;
}

// --- Kernel 1: segment boundaries via binary search + init encoded seg-max ---
__global__ void sg_init_kernel(const int* __restrict__ idx, int N,
                               unsigned* __restrict__ smax_enc,
                               int* __restrict__ start) {
  int t = threadIdx.x;
  if (t <= BMAX) {                 // start[b] = first i with idx[i] >= b; start[B]=N
    int lo = 0, hi = N;
    while (lo < hi) { int mid = (lo + hi) >> 1; if (idx[mid] < t) lo = mid + 1; else hi = mid; }
    start[t] = lo;
  }
  if (t < BMAX * KCLS) smax_enc[t] = enc_f32(-INFINITY);
}

// --- Kernel 2: per-(segment,class) max via monotonic-uint atomicMax (deterministic) ---
__global__ void sg_segmax_kernel(const float* __restrict__ probs,
                                 const int* __restrict__ idx, int N,
                                 unsigned* __restrict__ smax_enc) {
  int n = blockIdx.x * blockDim.x + threadIdx.x;
  if (n >= N) return;
  int b = idx[n];
  const float* p = probs + (size_t)n * KCLS;
  unsigned* s = smax_enc + b * KCLS;
#pragma unroll
  for (int k = 0; k < KCLS; ++k) atomicMax(&s[k], enc_f32(SCALE_F * p[k]));
}

// --- Kernel 3: per-chunk partial softmax denominators (fixed-order reduce in block) ---
__global__ void sg_denom_kernel(const float* __restrict__ probs,
                                const int* __restrict__ start,
                                const unsigned* __restrict__ smax_enc,
                                int chunk_len, int nchunk,
                                float* __restrict__ denom_part) {
  __shared__ float red[8][KCLS];
  int b = blockIdx.y, ch = blockIdx.x;
  int t = threadIdx.x;
  int k = t % KCLS, g = t / KCLS;        // 160 threads: 8 groups x 20 classes
  int r0 = start[b] + ch * chunk_len;
  int r1 = min(start[b + 1], r0 + chunk_len);
  float sm = dec_f32(smax_enc[b * KCLS + k]);
  if (!finitef_bits(sm)) sm = 0.0f;
  float acc = 0.0f;
  for (int n = r0 + g; n < r1; n += 8)
    acc += expf(SCALE_F * probs[(size_t)n * KCLS + k] - sm);
  red[g][k] = acc;
  __syncthreads();
  if (t < KCLS) {
    float s = 0.0f;
#pragma unroll
    for (int g2 = 0; g2 < 8; ++g2) s += red[g2][t];   // fixed order
    denom_part[((size_t)b * nchunk + ch) * KCLS + t] = s;
  }
}

// --- Kernel 4: reduce denominator partials over chunks (fixed order) ---
__global__ void sg_denom_reduce_kernel(const float* __restrict__ denom_part,
                                       float* __restrict__ denom, int nchunk) {
  int t = threadIdx.x;
  if (t >= BMAX * KCLS) return;
  int b = t / KCLS, k = t % KCLS;
  float s = 0.0f;
  for (int ch = 0; ch < nchunk; ++ch)
    s += denom_part[((size_t)b * nchunk + ch) * KCLS + k];
  denom[t] = s;
}

// --- Kernel 5: WMMA contraction. Block = 6 waves; wave w owns feature tile
//     [w*16, w*16+16). Two v8f accumulators per wave: classes 0-15 and 16-31
//     (20..31 padded with zero weights). ---
__global__ void __launch_bounds__(192)
sg_context_wmma_kernel(const float* __restrict__ probs,
                       const float* __restrict__ feats,
                       const int* __restrict__ start,
                       const unsigned* __restrict__ smax_enc,
                       const float* __restrict__ denom,
                       int N, int chunk_len, int nchunk,
                       float* __restrict__ ctx_part) {
  __shared__ float lds_w[SSTEP][WSTR];   // weights, 32 class slots used per row

  int b = blockIdx.y, ch = blockIdx.x;
  int tid  = threadIdx.x;
  int wave = tid >> 5;        // 0..5 -> feature tile
  int lane = tid & 31;
  int h = lane >> 4;          // lane half
  int m = lane & 15;
  int c = wave * 16 + m;      // this lane's feature column

  int r0 = start[b] + ch * chunk_len;
  int r1 = min(start[b + 1], r0 + chunk_len);

  v8f acc0 = {};              // classes 0..15
  v8f acc1 = {};              // classes 16..31 (16..19 valid)

  for (int base = r0; base < r1; base += SSTEP) {
    // Cooperative softmax-weight fill: probs read exactly once per point here.
    for (int i = tid; i < SSTEP * 32; i += 192) {
      int nl = i >> 5, k = i & 31;
      int n = base + nl;
      float w = 0.0f;
      if (n < r1 && k < KCLS) {
        float sm = dec_f32(smax_enc[b * KCLS + k]);
        if (!finitef_bits(sm)) sm = 0.0f;
        w = expf(SCALE_F * probs[(size_t)n * KCLS + k] - sm) / denom[b * KCLS + k];
      }
      lds_w[nl][k] = w;
    }
    __syncthreads();

#pragma unroll 4
    for (int s = 0; s < SSTEP / 4; ++s) {
      int nl0 = s * 4 + 2 * h;                    // lane-half's point rows
      // A (W^T): lane half h supplies K = 2h,2h+1; M = class = m (+kbase)
      float a0t0 = lds_w[nl0][m];
      float a1t0 = lds_w[nl0 + 1][m];
      float a0t1 = lds_w[nl0][16 + m];
      float a1t1 = lds_w[nl0 + 1][16 + m];
      // B (F): same row pattern; clamp rows past N (their weights are 0)
      long gr0 = base + nl0;     if (gr0 > (long)N - 1) gr0 = (long)N - 1;
      long gr1 = base + nl0 + 1; if (gr1 > (long)N - 1) gr1 = (long)N - 1;
      float b0 = feats[gr0 * CFEAT + c];
      float b1 = feats[gr1 * CFEAT + c];
      v2f A0 = {a0t0, a1t0};
      v2f A1 = {a0t1, a1t1};
      v2f Bv = {b0, b1};
      // D = A(16x4) x B(4x16) + C, fp32 throughout
      acc0 = __builtin_amdgcn_wmma_f32_16x16x4_f32(false, A0, false, Bv,
                                                   (short)0, acc0, false, false);
      acc1 = __builtin_amdgcn_wmma_f32_16x16x4_f32(false, A1, false, Bv,
                                                   (short)0, acc1, false, false);
    }
    __syncthreads();
  }

  // Write this chunk's partial tile (always, incl. zeros for empty ranges —
  // the reduce kernel sums every chunk slot and ws is poisoned).
  float* outp = ctx_part + ((size_t)b * nchunk + ch) * 32 * CFEAT;
#pragma unroll
  for (int v = 0; v < 8; ++v) {
    int k0 = v + 8 * h;                                  // C/D layout: M = v + 8*half
    outp[(size_t)k0 * CFEAT + c]        = acc0[v];
    outp[(size_t)(16 + k0) * CFEAT + c] = acc1[v];
  }
}

// --- Kernel 6: reduce contraction partials over chunks (fixed order), emit [B,K,C] ---
__global__ void sg_final_reduce_kernel(const float* __restrict__ ctx_part,
                                       float* __restrict__ out, int nchunk) {
  int i = blockIdx.x * blockDim.x + threadIdx.x;
  if (i >= BMAX * KCLS * CFEAT) return;
  int b = i / (KCLS * CFEAT);
  int rem = i % (KCLS * CFEAT);
  int k = rem / CFEAT, c = rem % CFEAT;
  float s = 0.0f;
  for (int ch = 0; ch < nchunk; ++ch)
    s += ctx_part[(((size_t)b * nchunk + ch) * 32 + k) * CFEAT + c];
  out[i] = s;
}

extern "C" void kernel_launch(void* const* d_in, const int* in_sizes, int n_in,
                              void* d_out, int out_size, void* d_ws, size_t ws_size,
                              hipStream_t stream) {
  const float* feats = (const float*)d_in[0];
  const float* probs = (const float*)d_in[1];
  // d_in[2] = batch_size scalar (B=8, fixed by reference shapes)
  const int* idx = (const int*)d_in[3];
  float* out = (float*)d_out;

  int N = in_sizes[0] / CFEAT;   // 131072

  // Workspace layout
  char* ws = (char*)d_ws;
  int*      start      = (int*)ws;                 // 16 ints
  unsigned* smax_enc   = (unsigned*)(ws + 64);     // B*K = 160
  float*    denom      = (float*)(ws + 1024);      // B*K = 160
  float*    denom_part = (float*)(ws + 4096);      // <= 64*B*K floats (40KB)
  float*    ctx_part   = (float*)(ws + 65536);     // nchunk*B*32*96 floats

  const size_t per_chunk = (size_t)BMAX * 32 * CFEAT * sizeof(float);  // 98304 B
  int nchunk = 1;
  if (ws_size > (size_t)65536 + per_chunk) {
    size_t q = (ws_size - 65536) / per_chunk;
    nchunk = (q < 64) ? (int)q : 64;
  }
  int chunk_len = ((N + nchunk - 1) / nchunk + SSTEP - 1) / SSTEP * SSTEP;

  sg_init_kernel<<<1, 256, 0, stream>>>(idx, N, smax_enc, start);
  sg_segmax_kernel<<<(N + 255) / 256, 256, 0, stream>>>(probs, idx, N, smax_enc);

  dim3 grid(nchunk, BMAX);
  sg_denom_kernel<<<grid, 160, 0, stream>>>(probs, start, smax_enc,
                                            chunk_len, nchunk, denom_part);
  sg_denom_reduce_kernel<<<1, 256, 0, stream>>>(denom_part, denom, nchunk);
  sg_context_wmma_kernel<<<grid, 192, 0, stream>>>(probs, feats, start, smax_enc,
                                                   denom, N, chunk_len, nchunk,
                                                   ctx_part);
  sg_final_reduce_kernel<<<(BMAX * KCLS * CFEAT + 255) / 256, 256, 0, stream>>>(
      ctx_part, out, nchunk);
}